// CGCNN_84542136254787
// MI455X (gfx1250) — compile-verified
//
#include <hip/hip_runtime.h>
#include <math.h>

#define N_NODES   100000
#define N_EDGES   3200000
#define N_GRAPHS  64
#define FEAT      128
#define NCLASS    144

#define NODES_PER_BLOCK 128
#define SLOTS 10
#define NEG_MAP 0x007FFFFFu   // order-preserving map of -inf

typedef __attribute__((ext_vector_type(2))) float v2f;
typedef __attribute__((ext_vector_type(8))) float v8f;

// Order-preserving float <-> uint mapping so we can use integer atomicMax.
__device__ __forceinline__ unsigned fmap(float f) {
    unsigned b = __float_as_uint(f);
    return (b & 0x80000000u) ? ~b : (b | 0x80000000u);
}
__device__ __forceinline__ float funmap(unsigned u) {
    unsigned b = (u & 0x80000000u) ? (u & 0x7FFFFFFFu) : ~u;
    return __uint_as_float(b);
}

// ---------------- init kernels ----------------
__global__ void cg_init_pmax(unsigned* __restrict__ pmax, int n) {
    int i = blockIdx.x * blockDim.x + threadIdx.x;
    if (i < n) pmax[i] = NEG_MAP;
}
__global__ void cg_zero_f32(float* __restrict__ p, int n) {
    int i = blockIdx.x * blockDim.x + threadIdx.x;
    if (i < n) p[i] = 0.0f;
}

// ---------------- edge kernel: gather, gate*core, scatter-add ----------------
__global__ void cg_edge(const float* __restrict__ x,
                        const int*   __restrict__ ei,     // [2, E] int32
                        const float* __restrict__ ea,     // [E, 4]
                        const float* __restrict__ Wf,     // [3,10]
                        const float* __restrict__ bf,     // [3]
                        const float* __restrict__ Ws,     // [3,10]
                        const float* __restrict__ bs,     // [3]
                        float* __restrict__ agg) {        // [N,3]
    __shared__ float sWf[30], sWs[30], sb[6];
    int tid = threadIdx.x;
    if (tid < 30) { sWf[tid] = Wf[tid]; sWs[tid] = Ws[tid]; }
    if (tid < 3)  { sb[tid] = bf[tid]; sb[3 + tid] = bs[tid]; }
    __syncthreads();

    int e = blockIdx.x * blockDim.x + tid;
    if (e >= N_EDGES) return;

    int src = ei[e];
    int dst = ei[N_EDGES + e];

    float z[10];
    z[0] = x[dst * 3 + 0]; z[1] = x[dst * 3 + 1]; z[2] = x[dst * 3 + 2];
    z[3] = x[src * 3 + 0]; z[4] = x[src * 3 + 1]; z[5] = x[src * 3 + 2];
    float4 a4 = ((const float4*)ea)[e];
    z[6] = a4.x; z[7] = a4.y; z[8] = a4.z; z[9] = a4.w;

#pragma unroll
    for (int c = 0; c < 3; ++c) {
        float vf = sb[c], vs = sb[3 + c];
#pragma unroll
        for (int k = 0; k < 10; ++k) {
            vf = fmaf(z[k], sWf[c * 10 + k], vf);
            vs = fmaf(z[k], sWs[c * 10 + k], vs);
        }
        float gate = 1.0f / (1.0f + expf(-vf));
        float core = fmaxf(vs, 0.0f) + log1pf(expf(-fabsf(vs)));  // stable softplus
        atomicAdd(&agg[dst * 3 + c], gate * core);
    }
}

// ---------------- node kernel: residual + (3->128 linear via WMMA) + segment-max ----------------
// Block = 256 threads = 8 waves; each wave owns 16 nodes; 8 WMMA col-tiles cover 128 outputs.
__global__ void cg_node(const float* __restrict__ xin,
                        const float* __restrict__ agg,
                        const int*   __restrict__ batch,
                        const float* __restrict__ Wlin,   // [128,3]
                        const float* __restrict__ blin,   // [128]
                        float* __restrict__ xout,         // [N,3]
                        unsigned* __restrict__ pmax) {    // [64,128] mapped uints
    __shared__ float    lx[NODES_PER_BLOCK][4];
    __shared__ unsigned smax[SLOTS * FEAT];
    __shared__ int      sg0, sspan;

    int tid  = threadIdx.x;
    int base = blockIdx.x * NODES_PER_BLOCK;
    int nvalid = N_NODES - base;
    if (nvalid > NODES_PER_BLOCK) nvalid = NODES_PER_BLOCK;

    // Stage A: residual x_new = x + agg -> global + LDS (K padded to 4 with zeros)
    for (int idx = tid; idx < NODES_PER_BLOCK * 3; idx += blockDim.x) {
        int ln = idx / 3, k = idx % 3;
        float v = 0.0f;
        if (ln < nvalid) {
            int n = base + ln;
            v = xin[n * 3 + k] + agg[n * 3 + k];
            xout[n * 3 + k] = v;
        }
        lx[ln][k] = v;
    }
    for (int idx = tid; idx < NODES_PER_BLOCK; idx += blockDim.x) lx[idx][3] = 0.0f;
    for (int idx = tid; idx < SLOTS * FEAT; idx += blockDim.x) smax[idx] = NEG_MAP;
    if (tid == 0) {
        int g0 = batch[base];
        int g1 = batch[base + nvalid - 1];
        sg0 = g0; sspan = g1 - g0 + 1;   // batch is sorted -> contiguous range
    }
    __syncthreads();

    int g0 = sg0, span = sspan;
    bool direct = (span > SLOTS);       // pathological fallback

    int wave = tid >> 5, lane = tid & 31;
    int half = lane >> 4, lm = lane & 15;
    int rowbase = wave * 16;

    // A-matrix (16x4 f32): lanes 0-15 -> (K0,K1), lanes 16-31 -> (K2,K3=0), M = lm
    v2f a;
    a.x = half ? lx[rowbase + lm][2] : lx[rowbase + lm][0];
    a.y = half ? 0.0f                : lx[rowbase + lm][1];

    // batch id per output row r (C-matrix: row M = half ? r+8 : r, col = lm)
    int  bg[8]; bool bv[8];
#pragma unroll
    for (int r = 0; r < 8; ++r) {
        int m  = half ? (r + 8) : r;
        int ln = rowbase + m;
        bv[r] = (ln < nvalid);
        bg[r] = bv[r] ? batch[base + ln] : 0;
    }

#pragma unroll
    for (int t = 0; t < 8; ++t) {
        int j = t * 16 + lm;
        // B-matrix (4x16 f32): lanes 0-15 -> rows (K0,K1), lanes 16-31 -> rows (K2,K3=0), N = lm
        v2f b;
        b.x = half ? Wlin[j * 3 + 2] : Wlin[j * 3 + 0];
        b.y = half ? 0.0f            : Wlin[j * 3 + 1];
        v8f c = {};
        c = __builtin_amdgcn_wmma_f32_16x16x4_f32(
                /*neg_a=*/false, a, /*neg_b=*/false, b,
                /*c_mod=*/(short)0, c, /*reuse_a=*/false, /*reuse_b=*/false);
        float bias = blin[j];
#pragma unroll
        for (int r = 0; r < 8; ++r) {
            if (!bv[r]) continue;
            unsigned mv = fmap(c[r] + bias);
            if (direct) atomicMax(&pmax[bg[r] * FEAT + j], mv);
            else        atomicMax(&smax[(bg[r] - g0) * FEAT + j], mv);
        }
    }
    __syncthreads();

    if (!direct) {
        int total = span * FEAT;
        for (int idx = tid; idx < total; idx += blockDim.x) {
            unsigned v = smax[idx];
            if (v != NEG_MAP)
                atomicMax(&pmax[(g0 + idx / FEAT) * FEAT + (idx & (FEAT - 1))], v);
        }
    }
}

// ---------------- classifier: (sum of 3 segment-maxes) @ W_cls^T + b_cls ----------------
__global__ void cg_cls(const unsigned* __restrict__ pmax,  // [3,64,128] mapped
                       const float* __restrict__ Wcls,     // [144,128]
                       const float* __restrict__ bcls,     // [144]
                       float* __restrict__ out) {          // [64,144]
    int idx = blockIdx.x * blockDim.x + threadIdx.x;
    if (idx >= N_GRAPHS * NCLASS) return;
    int g = idx / NCLASS, c = idx % NCLASS;
    float acc = bcls[c];
#pragma unroll 4
    for (int j = 0; j < FEAT; ++j) {
        float p = funmap(pmax[0 * N_GRAPHS * FEAT + g * FEAT + j])
                + funmap(pmax[1 * N_GRAPHS * FEAT + g * FEAT + j])
                + funmap(pmax[2 * N_GRAPHS * FEAT + g * FEAT + j]);
        acc = fmaf(p, Wcls[c * FEAT + j], acc);
    }
    out[g * NCLASS + c] = acc;
}

extern "C" void kernel_launch(void* const* d_in, const int* in_sizes, int n_in,
                              void* d_out, int out_size, void* d_ws, size_t ws_size,
                              hipStream_t stream) {
    const float* x     = (const float*)d_in[0];
    const int*   ei    = (const int*)  d_in[1];
    const float* ea    = (const float*)d_in[2];
    const int*   batch = (const int*)  d_in[3];
    const float* Wf[3] = {(const float*)d_in[4],  (const float*)d_in[8],  (const float*)d_in[12]};
    const float* bf[3] = {(const float*)d_in[5],  (const float*)d_in[9],  (const float*)d_in[13]};
    const float* Ws[3] = {(const float*)d_in[6],  (const float*)d_in[10], (const float*)d_in[14]};
    const float* bs[3] = {(const float*)d_in[7],  (const float*)d_in[11], (const float*)d_in[15]};
    const float* Wlin  = (const float*)d_in[16];
    const float* blin  = (const float*)d_in[17];
    const float* Wcls  = (const float*)d_in[18];
    const float* bcls  = (const float*)d_in[19];

    float* ws = (float*)d_ws;
    float*    xbuf0 = ws;                       // N*3
    float*    xbuf1 = ws + 3 * N_NODES;         // N*3
    float*    agg   = ws + 6 * N_NODES;         // N*3
    unsigned* pmax  = (unsigned*)(ws + 9 * N_NODES); // 3*64*128

    const int PMAXN = 3 * N_GRAPHS * FEAT;
    cg_init_pmax<<<(PMAXN + 255) / 256, 256, 0, stream>>>(pmax, PMAXN);

    const float* xin = x;
    float* xout = xbuf0;
    for (int L = 0; L < 3; ++L) {
        cg_zero_f32<<<(3 * N_NODES + 255) / 256, 256, 0, stream>>>(agg, 3 * N_NODES);
        cg_edge<<<(N_EDGES + 255) / 256, 256, 0, stream>>>(
            xin, ei, ea, Wf[L], bf[L], Ws[L], bs[L], agg);
        cg_node<<<(N_NODES + NODES_PER_BLOCK - 1) / NODES_PER_BLOCK, 256, 0, stream>>>(
            xin, agg, batch, Wlin, blin, xout, pmax + (size_t)L * N_GRAPHS * FEAT);
        xin  = xout;
        xout = (L == 0) ? xbuf1 : xbuf0;
    }

    cg_cls<<<(N_GRAPHS * NCLASS + 255) / 256, 256, 0, stream>>>(pmax, Wcls, bcls, (float*)d_out);
}